// RenderMesh_89369679495521
// MI455X (gfx1250) — compile-verified
//
#include <hip/hip_runtime.h>
#include <hip/hip_bf16.h>
#include <stdint.h>

#define IMGW 256
#define NPIX (IMGW*IMGW)
#define EPSF 1e-8f

typedef __attribute__((ext_vector_type(2))) float v2f;
typedef __attribute__((ext_vector_type(8))) float v8f;

// ---------------- K1: vertex transform + projection ----------------
__global__ void __launch_bounds__(256) k_vertex(
    const float* __restrict__ verts, const float* __restrict__ tm,
    const float* __restrict__ fl,
    float* __restrict__ xv, float* __restrict__ yv, float* __restrict__ zv, int V)
{
  int i = blockIdx.x * blockDim.x + threadIdx.x;
  if (i >= V) return;
  float a = verts[3*i+0], b = verts[3*i+1], c = verts[3*i+2];
  // v_view[j] = sum_i v[i]*R[i][j] + T[j];  R[i][j]=tm[i*4+j], T[j]=tm[j*4+3]
  float vx = a*tm[0] + b*tm[4] + c*tm[8]  + tm[3];
  float vy = a*tm[1] + b*tm[5] + c*tm[9]  + tm[7];
  float vz = a*tm[2] + b*tm[6] + c*tm[10] + tm[11];
  float zc = fmaxf(vz, EPSF);
  float fx = 2.0f * fl[0] / (float)IMGW;
  float fy = 2.0f * fl[1] / (float)IMGW;
  xv[i] = fx * vx / zc;
  yv[i] = fy * vy / zc;
  zv[i] = vz;
}

// ---------------- K2: face normals (original vertices) ----------------
__global__ void __launch_bounds__(256) k_facenorm(
    const float* __restrict__ verts, const int* __restrict__ faces,
    float* __restrict__ fnx, float* __restrict__ fny, float* __restrict__ fnz, int F)
{
  int f = blockIdx.x * blockDim.x + threadIdx.x;
  if (f >= F) return;
  int i0 = faces[3*f+0], i1 = faces[3*f+1], i2 = faces[3*f+2];
  float ax = verts[3*i1+0] - verts[3*i0+0];
  float ay = verts[3*i1+1] - verts[3*i0+1];
  float az = verts[3*i1+2] - verts[3*i0+2];
  float bx = verts[3*i2+0] - verts[3*i0+0];
  float by = verts[3*i2+1] - verts[3*i0+1];
  float bz = verts[3*i2+2] - verts[3*i0+2];
  fnx[f] = ay*bz - az*by;
  fny[f] = az*bx - ax*bz;
  fnz[f] = ax*by - ay*bx;
}

// ---------------- K3: vertex normals (deterministic LDS-staged scan) ----------------
#define VN_CHUNK 256
__global__ void __launch_bounds__(VN_CHUNK) k_vnorm(
    const int* __restrict__ faces,
    const float* __restrict__ fnx, const float* __restrict__ fny, const float* __restrict__ fnz,
    float* __restrict__ vnx, float* __restrict__ vny, float* __restrict__ vnz, int V, int F)
{
  __shared__ int   sf0[VN_CHUNK], sf1[VN_CHUNK], sf2[VN_CHUNK];
  __shared__ float sn0[VN_CHUNK], sn1[VN_CHUNK], sn2[VN_CHUNK];
  int v = blockIdx.x * blockDim.x + threadIdx.x;
  float ax = 0.0f, ay = 0.0f, az = 0.0f;
  for (int c0 = 0; c0 < F; c0 += VN_CHUNK) {
    __syncthreads();
    int t = threadIdx.x;
    int fi = c0 + t;
    if (fi < F) {
      sf0[t] = faces[3*fi+0]; sf1[t] = faces[3*fi+1]; sf2[t] = faces[3*fi+2];
      sn0[t] = fnx[fi];       sn1[t] = fny[fi];       sn2[t] = fnz[fi];
    }
    __syncthreads();
    int cnt = F - c0; if (cnt > VN_CHUNK) cnt = VN_CHUNK;
    if (v < V) {
      for (int j = 0; j < cnt; ++j) {
        int mc = (sf0[j] == v) + (sf1[j] == v) + (sf2[j] == v);
        if (mc) {
          float s = (float)mc;
          ax += s * sn0[j]; ay += s * sn1[j]; az += s * sn2[j];
        }
      }
    }
  }
  if (v < V) {
    float nrm = sqrtf(ax*ax + ay*ay + az*az) + EPSF;
    vnx[v] = ax / nrm; vny[v] = ay / nrm; vnz[v] = az / nrm;
  }
}

// ---------------- K4: per-triangle affine setup ----------------
// tri[f*16 + 0..15] = [C0,Bx0,By0,rz0 | C1,Bx1,By1,rz1 | C2,Bx2,By2,rz2 | Cz,Bxz,Byz,0]
// bK(P,Q) = CK + BxK*P + ByK*Q  (already divided by denom); invz(P,Q) = Cz + Bxz*P + Byz*Q
__global__ void __launch_bounds__(256) k_setup(
    const float* __restrict__ xv, const float* __restrict__ yv, const float* __restrict__ zv,
    const int* __restrict__ faces, float* __restrict__ tri, int F, int Fp)
{
  int f = blockIdx.x * blockDim.x + threadIdx.x;
  if (f >= Fp) return;
  float q[16];
  bool valid = false;
  if (f < F) {
    int i0 = faces[3*f+0], i1 = faces[3*f+1], i2 = faces[3*f+2];
    float x0 = xv[i0], x1 = xv[i1], x2 = xv[i2];
    float y0 = yv[i0], y1 = yv[i1], y2 = yv[i2];
    float z0 = zv[i0], z1 = zv[i1], z2 = zv[i2];
    float area = (x1-x0)*(y2-y0) - (x2-x0)*(y1-y0);
    valid = (fabsf(area) > EPSF) && (z0 > EPSF) && (z1 > EPSF) && (z2 > EPSF);
    float invd = 1.0f / (valid ? area : 1.0f);
    float rz0 = 1.0f / fmaxf(z0, EPSF);
    float rz1 = 1.0f / fmaxf(z1, EPSF);
    float rz2 = 1.0f / fmaxf(z2, EPSF);
    float C0 = (x1*y2 - x2*y1)*invd, Bx0 = (y1-y2)*invd, By0 = (x2-x1)*invd;
    float C1 = (x2*y0 - x0*y2)*invd, Bx1 = (y2-y0)*invd, By1 = (x0-x2)*invd;
    float C2 = (x0*y1 - x1*y0)*invd, Bx2 = (y0-y1)*invd, By2 = (x1-x0)*invd;
    q[0]=C0;  q[1]=Bx0;  q[2]=By0;  q[3]=rz0;
    q[4]=C1;  q[5]=Bx1;  q[6]=By1;  q[7]=rz1;
    q[8]=C2;  q[9]=Bx2;  q[10]=By2; q[11]=rz2;
    q[12]=C0*rz0 + C1*rz1 + C2*rz2;
    q[13]=Bx0*rz0 + Bx1*rz1 + Bx2*rz2;
    q[14]=By0*rz0 + By1*rz1 + By2*rz2;
    q[15]=0.0f;
  }
  if (!valid) {
    #pragma unroll
    for (int k = 0; k < 16; ++k) q[k] = 0.0f;
    q[0] = q[4] = q[8] = -1.0f;   // b0=b1=b2=-1 -> never inside
  }
  float4* o = (float4*)(tri + (size_t)f * 16);
  o[0] = make_float4(q[0],  q[1],  q[2],  q[3]);
  o[1] = make_float4(q[4],  q[5],  q[6],  q[7]);
  o[2] = make_float4(q[8],  q[9],  q[10], q[11]);
  o[3] = make_float4(q[12], q[13], q[14], q[15]);
}

// ---------------- K5: WMMA rasterizer with LDS-staged triangle chunks ----------------
// One wave = 16 consecutive pixels (rows M); block (8 waves) shares each 64-face chunk
// via LDS. Chunk is software-pipelined: global b128 load of chunk c+1 held in a
// register while chunk c is consumed from LDS; one barrier per chunk.
// LDS face stride padded to 20 floats -> the (n*20 + q*4 + hi*2) ds_load_b64 pattern
// touches 64 distinct banks across wave32 (conflict-free).
#define CH_FACES 64
#define LDS_STRIDE 20
__global__ void __launch_bounds__(256) k_raster(
    const float* __restrict__ tri, float* __restrict__ out_iz,
    int* __restrict__ out_f, int nchunks)
{
  __shared__ float sbuf[2][CH_FACES * LDS_STRIDE];

  const int lane = (int)(threadIdx.x & 31);
  const int gw   = (int)((blockIdx.x * blockDim.x + threadIdx.x) >> 5);
  const int n    = lane & 15;          // matrix column / A row within half
  const int hi   = lane >> 4;          // lane half: K split

  // pixel NDC coords for A-fragment row M = n
  int p0i = gw * 16 + n;
  int h = p0i >> 8, w = p0i & (IMGW - 1);
  float P = 1.0f - 2.0f * ((float)w + 0.5f) / (float)IMGW;
  float Q = 1.0f - 2.0f * ((float)h + 0.5f) / (float)IMGW;

  // A 16x4 layout: lanes 0-15 hold K=0,1 ; lanes 16-31 hold K=2,3
  v2f afrag;
  afrag.x = (hi == 0) ? 1.0f : Q;
  afrag.y = (hi == 0) ? P    : 0.0f;

  float best_iz[8];
  int   best_f[8];
  #pragma unroll
  for (int r = 0; r < 8; ++r) { best_iz[r] = 0.0f; best_f[r] = -1; }

  v8f zero = {0.0f,0.0f,0.0f,0.0f,0.0f,0.0f,0.0f,0.0f};

  // staging role of this thread: face fi (0..63), quarter part (0..3) of its 16 floats
  const int t    = (int)threadIdx.x;
  const int fi   = t >> 2;
  const int part = t & 3;
  const float4* __restrict__ tri4 = (const float4*)tri;   // 4 float4 per face

  // prologue: stage chunk 0
  float4 reg = tri4[t];
  *(float4*)(&sbuf[0][fi * LDS_STRIDE + part * 4]) = reg;

  for (int c = 0; c < nchunks; ++c) {
    if (c + 1 < nchunks) reg = tri4[(size_t)(c + 1) * 256 + t];
    __syncthreads();   // chunk c staged and all waves past previous reads
    const float* sb = sbuf[c & 1];
    #pragma unroll
    for (int j = 0; j < 4; ++j) {
      // B 4x16: lanes 0-15 -> (K0,K1)=(C,Bx); lanes 16-31 -> (K2,K3)=(By,pad)
      const float* tb = sb + (j * 16 + n) * LDS_STRIDE + hi * 2;
      v2f bf0 = *(const v2f*)(tb + 0);
      v2f bf1 = *(const v2f*)(tb + 4);
      v2f bf2 = *(const v2f*)(tb + 8);
      v2f bfz = *(const v2f*)(tb + 12);

      v8f d0 = __builtin_amdgcn_wmma_f32_16x16x4_f32(false, afrag, false, bf0, (short)0, zero, false, false);
      v8f d1 = __builtin_amdgcn_wmma_f32_16x16x4_f32(false, afrag, false, bf1, (short)0, zero, false, false);
      v8f d2 = __builtin_amdgcn_wmma_f32_16x16x4_f32(false, afrag, false, bf2, (short)0, zero, false, false);
      v8f dz = __builtin_amdgcn_wmma_f32_16x16x4_f32(false, afrag, false, bfz, (short)0, zero, false, false);

      int fcol = c * CH_FACES + j * 16 + n;   // this lane's D column = triangle id
      #pragma unroll
      for (int r = 0; r < 8; ++r) {
        float mn = fminf(d0[r], fminf(d1[r], d2[r]));
        bool upd = (mn >= 0.0f) && (dz[r] > best_iz[r]);
        if (upd) { best_iz[r] = dz[r]; best_f[r] = fcol; }
      }
    }
    if (c + 1 < nchunks)
      *(float4*)(&sbuf[(c + 1) & 1][fi * LDS_STRIDE + part * 4]) = reg;
  }

  // Reduce across the 16 columns (lanes within each half-wave).
  // Key: (invz asc) major, (smaller face id wins ties) minor.
  #pragma unroll
  for (int r = 0; r < 8; ++r) {
    unsigned long long key =
        ((unsigned long long)__float_as_uint(best_iz[r]) << 32) |
        (unsigned long long)(0xFFFFFFFFu - (unsigned)best_f[r]);
    #pragma unroll
    for (int msk = 1; msk < 16; msk <<= 1) {
      unsigned long long o = __shfl_xor(key, msk, 32);
      key = (o > key) ? o : key;
    }
    if (n == 0) {
      int M = (hi << 3) + r;                 // D rows: lanes<16 -> M=r, lanes>=16 -> M=8+r
      int pix = gw * 16 + M;
      out_iz[pix] = __uint_as_float((unsigned)(key >> 32));
      out_f[pix]  = (int)(0xFFFFFFFFu - (unsigned)(key & 0xFFFFFFFFull));
    }
  }
}

// ---------------- K6: Phong shading ----------------
__global__ void __launch_bounds__(256) k_shade(
    const float* __restrict__ verts,
    const float* __restrict__ vnx, const float* __restrict__ vny, const float* __restrict__ vnz,
    const int* __restrict__ faces, const float* __restrict__ tri,
    const float* __restrict__ tm,
    const float* __restrict__ out_iz, const int* __restrict__ out_f,
    float* __restrict__ out)
{
  int p = blockIdx.x * blockDim.x + threadIdx.x;
  if (p >= NPIX) return;
  int f = out_f[p];
  float rgb = 0.0f, alpha = 0.0f;
  if (f >= 0) {
    int h = p >> 8, w = p & (IMGW - 1);
    float P = 1.0f - 2.0f * ((float)w + 0.5f) / (float)IMGW;
    float Q = 1.0f - 2.0f * ((float)h + 0.5f) / (float)IMGW;
    const float* tb = tri + (size_t)f * 16;
    float b0 = tb[0] + tb[1]*P + tb[2]*Q;
    float b1 = tb[4] + tb[5]*P + tb[6]*Q;
    float t0 = b0 * tb[3];
    float t1 = b1 * tb[7];
    float zpix = 1.0f / fmaxf(out_iz[p], EPSF);
    float w0 = t0 * zpix, w1 = t1 * zpix, w2 = 1.0f - w0 - w1;

    int i0 = faces[3*f+0], i1 = faces[3*f+1], i2 = faces[3*f+2];
    float Px = w0*verts[3*i0+0] + w1*verts[3*i1+0] + w2*verts[3*i2+0];
    float Py = w0*verts[3*i0+1] + w1*verts[3*i1+1] + w2*verts[3*i2+1];
    float Pz = w0*verts[3*i0+2] + w1*verts[3*i1+2] + w2*verts[3*i2+2];

    float Nx = w0*vnx[i0] + w1*vnx[i1] + w2*vnx[i2];
    float Ny = w0*vny[i0] + w1*vny[i1] + w2*vny[i2];
    float Nz = w0*vnz[i0] + w1*vnz[i1] + w2*vnz[i2];
    float nn = sqrtf(Nx*Nx + Ny*Ny + Nz*Nz) + EPSF;
    Nx /= nn; Ny /= nn; Nz /= nn;

    // cam[k] = -sum_j T[j]*R[k][j]
    float T0 = tm[3], T1 = tm[7], T2 = tm[11];
    float cx = -(T0*tm[0] + T1*tm[1]  + T2*tm[2]);
    float cy = -(T0*tm[4] + T1*tm[5]  + T2*tm[6]);
    float cz = -(T0*tm[8] + T1*tm[9]  + T2*tm[10]);

    float Lx = 0.0f - Px, Ly = 0.0f - Py, Lz = 3.0f - Pz;
    float ln = sqrtf(Lx*Lx + Ly*Ly + Lz*Lz) + EPSF;
    Lx /= ln; Ly /= ln; Lz /= ln;

    float Vx = cx - Px, Vy = cy - Py, Vz = cz - Pz;
    float vn_ = sqrtf(Vx*Vx + Vy*Vy + Vz*Vz) + EPSF;
    Vx /= vn_; Vy /= vn_; Vz /= vn_;

    float dNL = Nx*Lx + Ny*Ly + Nz*Lz;
    float ndl = fmaxf(dNL, 0.0f);
    float Rx = -Lx + 2.0f*dNL*Nx;
    float Ry = -Ly + 2.0f*dNL*Ny;
    float Rz = -Lz + 2.0f*dNL*Nz;
    float rv = fmaxf(Rx*Vx + Ry*Vy + Rz*Vz, 0.0f);
    float s = rv;
    s *= s; s *= s; s *= s; s *= s; s *= s; s *= s;   // rv^64
    float shade = 0.5f + 0.3f*ndl + 0.2f*s;
    rgb = shade * 255.0f;
    alpha = 1.0f;
  }
  out[0*NPIX + p] = rgb;
  out[1*NPIX + p] = rgb;
  out[2*NPIX + p] = rgb;
  out[3*NPIX + p] = alpha;   // alpha plane follows the 3 image planes
}

extern "C" void kernel_launch(void* const* d_in, const int* in_sizes, int n_in,
                              void* d_out, int out_size, void* d_ws, size_t ws_size,
                              hipStream_t stream) {
  const float* verts = (const float*)d_in[0];
  const float* tm    = (const float*)d_in[1];
  const float* fl    = (const float*)d_in[2];
  const int*   faces = (const int*)d_in[3];
  const int V  = in_sizes[0] / 3;
  const int F  = in_sizes[3] / 3;
  const int Fp = ((F + CH_FACES - 1) / CH_FACES) * CH_FACES;  // multiple of 64
  const int nchunks = Fp / CH_FACES;

  char* ws = (char*)d_ws;
  size_t off = 0;
  auto take = [&](size_t bytes) -> void* {
    void* ptr = (void*)(ws + off);
    off += (bytes + 255) & ~(size_t)255;
    return ptr;
  };
  float* tri  = (float*)take((size_t)Fp * 16 * sizeof(float));
  float* xv   = (float*)take((size_t)V * sizeof(float));
  float* yv   = (float*)take((size_t)V * sizeof(float));
  float* zv   = (float*)take((size_t)V * sizeof(float));
  float* fnx  = (float*)take((size_t)F * sizeof(float));
  float* fny  = (float*)take((size_t)F * sizeof(float));
  float* fnz  = (float*)take((size_t)F * sizeof(float));
  float* vnx  = (float*)take((size_t)V * sizeof(float));
  float* vny  = (float*)take((size_t)V * sizeof(float));
  float* vnz  = (float*)take((size_t)V * sizeof(float));
  float* piz  = (float*)take((size_t)NPIX * sizeof(float));
  int*   pf   = (int*)  take((size_t)NPIX * sizeof(int));
  (void)ws_size; (void)n_in; (void)out_size;

  const int B = 256;
  k_vertex  <<<(V + B - 1) / B, B, 0, stream>>>(verts, tm, fl, xv, yv, zv, V);
  k_facenorm<<<(F + B - 1) / B, B, 0, stream>>>(verts, faces, fnx, fny, fnz, F);
  k_vnorm   <<<(V + B - 1) / B, B, 0, stream>>>(faces, fnx, fny, fnz, vnx, vny, vnz, V, F);
  k_setup   <<<(Fp + B - 1) / B, B, 0, stream>>>(xv, yv, zv, faces, tri, F, Fp);

  // one wave per 16 pixels -> NPIX/16 waves -> *32 threads
  int raster_threads = (NPIX / 16) * 32;
  k_raster  <<<raster_threads / B, B, 0, stream>>>(tri, piz, pf, nchunks);

  k_shade   <<<(NPIX + B - 1) / B, B, 0, stream>>>(verts, vnx, vny, vnz, faces, tri, tm,
                                                   piz, pf, (float*)d_out);
}